// ENO_28561532519089
// MI455X (gfx1250) — compile-verified
//
#include <hip/hip_runtime.h>

typedef __attribute__((ext_vector_type(2))) float v2f;
typedef __attribute__((ext_vector_type(8))) float v8f;

#define IN_DIM 258

// ---------------------------------------------------------------------------
// Precompute base[r][j] = Wo1[j, 2:258] . u0[r, :] + bo1[j]   (32 x 256)
// ---------------------------------------------------------------------------
__global__ void eno_base_kernel(const float* __restrict__ Wo1,
                                const float* __restrict__ bo1,
                                const float* __restrict__ u0,
                                float* __restrict__ base) {
  const int r = blockIdx.x;
  const int j = threadIdx.x;
  const float* w = Wo1 + j * IN_DIM + 2;
  const float* u = u0 + r * 256;
  float s = bo1[j];
#pragma unroll 8
  for (int k = 0; k < 256; ++k) s = fmaf(w[k], u[k], s);
  base[r * 256 + j] = s;
}

// ---------------------------------------------------------------------------
// f32 WMMA GEMM: D[256 x (NV*16)] = W(256x256)[ᵀ] @ Bls
// Wave `wave` computes rows [32*wave, 32*wave+32). Bls layout: [k][80], the
// NV variants at column offsets v*16. Fragments per CDNA5 f32 16x16x4 layout.
// ---------------------------------------------------------------------------
template <int NV, bool TRANS>
__device__ __forceinline__ void gemm_f32(const float* __restrict__ W,
                                         const float* __restrict__ Bls,
                                         v8f acc[2][NV], int wave, int lane) {
  const int col = lane & 15;
  const int kh = lane >> 4;
  const v8f zero = {0.f, 0.f, 0.f, 0.f, 0.f, 0.f, 0.f, 0.f};
#pragma unroll
  for (int mt = 0; mt < 2; ++mt)
#pragma unroll
    for (int v = 0; v < NV; ++v) acc[mt][v] = zero;

#pragma unroll 4
  for (int k0 = 0; k0 < 256; k0 += 4) {
    const int k = k0 + 2 * kh;  // this lane-half's K pair
    v2f b[NV];
#pragma unroll
    for (int v = 0; v < NV; ++v) {
      b[v].x = Bls[k * 80 + v * 16 + col];
      b[v].y = Bls[(k + 1) * 80 + v * 16 + col];
    }
#pragma unroll
    for (int mt = 0; mt < 2; ++mt) {
      const int row = 32 * wave + 16 * mt + col;
      v2f a;
      if (TRANS) {
        a.x = W[k * 256 + row];
        a.y = W[(k + 1) * 256 + row];
      } else {
        a = *(const v2f*)(W + row * 256 + k);
      }
#pragma unroll
      for (int v = 0; v < NV; ++v)
        acc[mt][v] = __builtin_amdgcn_wmma_f32_16x16x4_f32(
            false, a, false, b[v], (short)0, acc[mt][v], false, false);
    }
  }
}

// ---------------------------------------------------------------------------
// Main kernel: one block (256 thr = 8 waves) per 16 consecutive query points.
// ---------------------------------------------------------------------------
__global__ __launch_bounds__(256) void eno_main_kernel(
    const float* __restrict__ qx, const float* __restrict__ qt,
    const float* __restrict__ Wo1, const float* __restrict__ base,
    const float* __restrict__ Wo2, const float* __restrict__ bo2,
    const float* __restrict__ Wo3, const float* __restrict__ Wh1,
    const float* __restrict__ bh1, const float* __restrict__ Wh2,
    const float* __restrict__ bh2, const float* __restrict__ Wh3,
    float* __restrict__ out, int M, int COL) {
  __shared__ float Bls[256 * 80];      // GEMM B operand, 80 cols (5x16 max)
  __shared__ float R1s[3][256 * 16];   // r1, r1_x, r1_xx persisted
  __shared__ float red[5][16];         // u, u_x, u_t, u_xx, u_xxx per point
  __shared__ float hacc[2][16];        // dHu_dx, dHux_dxx accumulators

  const int tid = threadIdx.x;
  const int wave = tid >> 5;
  const int lane = tid & 31;
  const int col = lane & 15;
  const int kh = lane >> 4;
  const int g0 = blockIdx.x * 16;
  const float* bs = base + (g0 / COL) * 256;

  if (tid < 16) {
#pragma unroll
    for (int v = 0; v < 5; ++v) red[v][tid] = 0.f;
    hacc[0][tid] = 0.f;
    hacc[1][tid] = 0.f;
  }

  // ---- Stage 1: operator layer-1 tangent vectors -> Bls (5 variants) ----
  for (int idx = tid; idx < 256 * 16; idx += 256) {
    const int j = idx >> 4, p = idx & 15;
    const float x = qx[g0 + p], t = qt[g0 + p];
    const float wx = Wo1[j * IN_DIM + 0], wt = Wo1[j * IN_DIM + 1];
    const float a1 = fmaf(x, wx, fmaf(t, wt, bs[j]));
    const float h1 = tanhf(a1);
    const float s1 = 1.f - h1 * h1;
    float* B = Bls + j * 80;
    B[0 * 16 + p] = h1;                                        // -> a2
    B[1 * 16 + p] = s1 * wx;                                   // -> a2_x
    B[2 * 16 + p] = s1 * wt;                                   // -> a2_t
    B[3 * 16 + p] = -2.f * h1 * s1 * wx * wx;                  // -> a2_xx
    B[4 * 16 + p] = -2.f * s1 * (s1 - 2.f * h1 * h1) * wx * wx * wx;  // a2_xxx
  }
  __syncthreads();

  // ---- GEMM1: Wo2 @ {5 variants}; pointwise tanh chain; Wo3 dots ----
  {
    v8f acc[2][5];
    gemm_f32<5, false>(Wo2, Bls, acc, wave, lane);
    float pu = 0.f, pux = 0.f, put = 0.f, puxx = 0.f, puxxx = 0.f;
#pragma unroll
    for (int mt = 0; mt < 2; ++mt)
#pragma unroll
      for (int r = 0; r < 8; ++r) {
        const int j = 32 * wave + 16 * mt + 8 * kh + r;
        const float a2v = acc[mt][0][r] + bo2[j];
        const float a2x = acc[mt][1][r], a2t = acc[mt][2][r];
        const float a2xx = acc[mt][3][r], a2xxx = acc[mt][4][r];
        const float h2 = tanhf(a2v);
        const float s2 = 1.f - h2 * h2;
        const float w3 = Wo3[j];
        pu += w3 * h2;
        pux += w3 * (s2 * a2x);
        put += w3 * (s2 * a2t);
        puxx += w3 * (s2 * a2xx - 2.f * h2 * s2 * a2x * a2x);
        puxxx += w3 * (s2 * a2xxx - 6.f * h2 * s2 * a2x * a2xx -
                       2.f * s2 * (s2 - 2.f * h2 * h2) * a2x * a2x * a2x);
      }
    atomicAdd(&red[0][col], pu);
    atomicAdd(&red[1][col], pux);
    atomicAdd(&red[2][col], put);
    atomicAdd(&red[3][col], puxx);
    atomicAdd(&red[4][col], puxxx);
  }
  __syncthreads();

  // ---- Stage 2: Hamiltonian layer-1 (z = [u,u_x], x-derivative chains) ----
  for (int idx = tid; idx < 256 * 16; idx += 256) {
    const int j = idx >> 4, p = idx & 15;
    const float u = red[0][p], ux = red[1][p], uxx = red[3][p],
                uxxx = red[4][p];
    const float w0 = Wh1[2 * j], w1 = Wh1[2 * j + 1];
    const float b1 = fmaf(w0, u, fmaf(w1, ux, bh1[j]));
    const float b1x = fmaf(w0, ux, w1 * uxx);
    const float b1xx = fmaf(w0, uxx, w1 * uxxx);
    const float g1 = tanhf(b1);
    const float r1 = 1.f - g1 * g1;
    float* B = Bls + j * 80;
    B[p] = g1;                                           // -> b2
    B[16 + p] = r1 * b1x;                                // -> b2_x
    B[32 + p] = r1 * b1xx - 2.f * g1 * r1 * b1x * b1x;   // -> b2_xx
    R1s[0][j * 16 + p] = r1;
    R1s[1][j * 16 + p] = -2.f * g1 * r1 * b1x;
    R1s[2][j * 16 + p] =
        -2.f * r1 * ((r1 - 2.f * g1 * g1) * b1x * b1x + g1 * b1xx);
  }
  __syncthreads();

  // ---- GEMM2: Wh2 @ {g1, g1_x, g1_xx}; pointwise -> p, p_x, p_xx ----
  {
    v8f acc[2][3];
    gemm_f32<3, false>(Wh2, Bls, acc, wave, lane);
    __syncthreads();  // all B reads done before overwriting Bls
#pragma unroll
    for (int mt = 0; mt < 2; ++mt)
#pragma unroll
      for (int r = 0; r < 8; ++r) {
        const int j = 32 * wave + 16 * mt + 8 * kh + r;
        const float b2 = acc[mt][0][r] + bh2[j];
        const float b2x = acc[mt][1][r], b2xx = acc[mt][2][r];
        const float g2 = tanhf(b2);
        const float r2 = 1.f - g2 * g2;
        const float r2x = -2.f * g2 * r2 * b2x;
        const float r2xx =
            -2.f * r2 * ((r2 - 2.f * g2 * g2) * b2x * b2x + g2 * b2xx);
        const float w3 = Wh3[j];
        float* B = Bls + j * 80;
        B[col] = r2 * w3;
        B[16 + col] = r2x * w3;
        B[32 + col] = r2xx * w3;
      }
  }
  __syncthreads();

  // ---- GEMM3: Wh2^T @ {p, p_x, p_xx}; combine with r1 chain; Wh1 dots ----
  {
    v8f acc[2][3];
    gemm_f32<3, true>(Wh2, Bls, acc, wave, lane);
    float hu = 0.f, hx = 0.f;
#pragma unroll
    for (int mt = 0; mt < 2; ++mt)
#pragma unroll
      for (int r = 0; r < 8; ++r) {
        const int j = 32 * wave + 16 * mt + 8 * kh + r;
        const float q = acc[mt][0][r], qx_ = acc[mt][1][r],
                    qxx = acc[mt][2][r];
        const float r1 = R1s[0][j * 16 + col];
        const float r1x = R1s[1][j * 16 + col];
        const float r1xx = R1s[2][j * 16 + col];
        const float mx = r1x * q + r1 * qx_;
        const float mxx = r1xx * q + 2.f * r1x * qx_ + r1 * qxx;
        hu += Wh1[2 * j] * mx;        // d/dx (dH/du)
        hx += Wh1[2 * j + 1] * mxx;   // d2/dx2 (dH/du_x)
      }
    atomicAdd(&hacc[0][col], hu);
    atomicAdd(&hacc[1][col], hx);
  }
  __syncthreads();

  if (tid < 16) {
    out[g0 + tid] = red[2][tid];                       // pred_dudt = u_t
    out[M + g0 + tid] = hacc[0][tid] - hacc[1][tid];   // H_eq
  }
}

// ---------------------------------------------------------------------------
extern "C" void kernel_launch(void* const* d_in, const int* in_sizes, int n_in,
                              void* d_out, int out_size, void* d_ws,
                              size_t ws_size, hipStream_t stream) {
  const float* qx = (const float*)d_in[0];
  const float* qt = (const float*)d_in[1];
  const float* u0 = (const float*)d_in[2];
  const float* Wo1 = (const float*)d_in[3];
  const float* bo1 = (const float*)d_in[4];
  const float* Wo2 = (const float*)d_in[5];
  const float* bo2 = (const float*)d_in[6];
  const float* Wo3 = (const float*)d_in[7];
  const float* Wh1 = (const float*)d_in[8];
  const float* bh1 = (const float*)d_in[9];
  const float* Wh2 = (const float*)d_in[10];
  const float* bh2 = (const float*)d_in[11];
  const float* Wh3 = (const float*)d_in[12];

  const int M = in_sizes[0];          // 131072
  const int Nrows = in_sizes[2] / 256;  // 32
  const int COL = M / Nrows;          // 4096

  float* base = (float*)d_ws;  // Nrows*256 floats

  eno_base_kernel<<<Nrows, 256, 0, stream>>>(Wo1, bo1, u0, base);
  eno_main_kernel<<<M / 16, 256, 0, stream>>>(qx, qt, Wo1, base, Wo2, bo2,
                                              Wo3, Wh1, bh1, Wh2, bh2, Wh3,
                                              (float*)d_out, M, COL);
}